// ControlValLoss_4672924418665
// MI455X (gfx1250) — compile-verified
//
#include <hip/hip_runtime.h>
#include <stdint.h>

// ---------------------------------------------------------------------------
// ControlValLoss on MI455X (gfx1250)
//
// Memory-bound streaming reduction: pred = 256 x 2048 x 206 f32 = 432 MB,
// output = 2 scalars. HBM floor @ 23.3 TB/s ~= 18.5 us. No matmul structure
// -> no WMMA; the CDNA5 win is data movement: coalesced NT b64 loads for the
// two argmax rows + async global->LDS staging (ASYNCcnt path) of the softmax
// row so its fetch overlaps the argmax work. VALU per wave trimmed (__expf,
// strict-> in-lane argmax, compile-time softmax group split) to stay under
// the ~300-cycle/wave bandwidth budget.
// ---------------------------------------------------------------------------

#define TOKENS     206
#define HALFTOK    101.0f          // (206-4)/2
#define IGNORE_IDX (-100)
#define Bdim       256
#define Sdim       682
#define Tdim       2048            // 3*682+2
#define NPAIR      103             // float2 pairs per 206-float row
#define WAVES      8               // waves per block (wave32)
#define BLOCK      256
#define NROWS      (Bdim * Sdim)   // 174592 (b,s) triples, one wave each
#define NBLK       (NROWS / WAVES) // 21824 blocks, exact fit

#define USE_ASYNC_LDS 1

typedef float v2f __attribute__((ext_vector_type(2)));

#define NEG_HUGE (-3.402823466e38f)

__device__ __forceinline__ v2f nt2(const float* p) {
    // rows are 824B (8B aligned) -> b64 is the widest safe load; NT because
    // the 432MB stream is touched exactly once (don't thrash the 192MB L2).
    return __builtin_nontemporal_load((const v2f*)p);
}

__device__ __forceinline__ float wave_sum(float v) {
#pragma unroll
    for (int off = 16; off; off >>= 1) v += __shfl_xor(v, off, 32);
    return v;
}

__device__ __forceinline__ float wave_max(float v) {
#pragma unroll
    for (int off = 16; off; off >>= 1) v = fmaxf(v, __shfl_xor(v, off, 32));
    return v;
}

// argmax over one 206-float row; first-occurrence tie-break like jnp.argmax.
__device__ __forceinline__ int row_argmax(const float* row, int lane) {
    float best = NEG_HUGE;
    int   bidx = 0x7FFFFFFF;
#pragma unroll
    for (int i = 0; i < 4; ++i) {
        int p = lane + 32 * i;
        if (p < NPAIR) {
            v2f v = nt2(row + 2 * p);
            // in-lane indices strictly increase, so strict '>' already keeps
            // the first occurrence; no index compare needed here.
            if (v.x > best) { best = v.x; bidx = 2 * p; }
            if (v.y > best) { best = v.y; bidx = 2 * p + 1; }
        }
    }
    // cross-lane: equal-value -> smaller-element-index wins
#pragma unroll
    for (int off = 16; off; off >>= 1) {
        float ob = __shfl_xor(best, off, 32);
        int   oi = __shfl_xor(bidx, off, 32);
        if (ob > best || (ob == best && oi < bidx)) { best = ob; bidx = oi; }
    }
    return bidx;
}

__device__ __forceinline__ float smooth_l1(float p, float t) {
    float d = p - t, ad = fabsf(d);
    return ad < 1.0f ? 0.5f * d * d : ad - 0.5f;
}

__global__ __launch_bounds__(BLOCK) void control_loss_partial(
    const float* __restrict__ pred,
    const float* __restrict__ gt_acc,
    const float* __restrict__ gt_steer,
    const int*  __restrict__ gt_rev,
    float* __restrict__ partial /* [NBLK][4] */)
{
    __shared__ float s_part[WAVES][4];
#if USE_ASYNC_LDS
    __shared__ float s_rev[WAVES][TOKENS + 2];   // per-wave staging slice
#endif

    const int lane = threadIdx.x & 31;
    const int wid  = threadIdx.x >> 5;
    const int row  = blockIdx.x * WAVES + wid;   // (b,s) triple id, exact fit
    const int b    = row / Sdim;
    const int s    = row - b * Sdim;
    const float* base = pred + ((size_t)b * Tdim + 3 * (size_t)s) * TOKENS;

#if USE_ASYNC_LDS
    // Kick off CDNA5 async copy of the reverse-logits row into this wave's
    // LDS slice; its fetch overlaps the two argmax rows below. LDS operand:
    // low 32 bits of the flat shared address == workgroup-relative LDS offset
    // (ISA 10.2 aperture rule), which is what the DS-address VGPR wants.
    {
        const float* rrow = base + 2 * TOKENS;
        unsigned lds_base = (unsigned)(uintptr_t)(&s_rev[wid][0]);
#pragma unroll
        for (int i = 0; i < 4; ++i) {
            int p = lane + 32 * i;
            if (p < NPAIR) {
                unsigned           laddr = lds_base + 8u * (unsigned)p;
                unsigned long long gaddr = (unsigned long long)(uintptr_t)(rrow + 2 * p);
                asm volatile("global_load_async_to_lds_b64 %0, %1, off th:TH_LOAD_NT"
                             :: "v"(laddr), "v"(gaddr) : "memory");
            }
        }
    }
#endif

    // ---- acc / steer: argmax rows straight from global (NT b64 loads) ----
    const int accIdx   = row_argmax(base, lane);
    const int steerIdx = row_argmax(base + TOKENS, lane);

    // ---- reverse row values into registers ----
    v2f r[4];
#if USE_ASYNC_LDS
    asm volatile("s_wait_asynccnt 0" ::: "memory");
#pragma unroll
    for (int i = 0; i < 4; ++i) {
        int p = lane + 32 * i;
        if (p < NPAIR) {
            r[i] = *(const v2f*)&s_rev[wid][2 * p];
        } else {
            r[i].x = NEG_HUGE; r[i].y = NEG_HUGE;
        }
    }
#else
    {
        const float* rrow = base + 2 * TOKENS;
#pragma unroll
        for (int i = 0; i < 4; ++i) {
            int p = lane + 32 * i;
            if (p < NPAIR) {
                r[i] = nt2(rrow + 2 * p);
            } else {
                r[i].x = NEG_HUGE; r[i].y = NEG_HUGE;
            }
        }
    }
#endif

    // softmax over 206, split-summed at element index 101. Group membership
    // per register slot is known at compile time except slot 1:
    //   slot0: e in [0,63]   -> A        slot1: e in [64,127] -> straddles
    //   slot2: e in [128,191]-> B        slot3: e in [192,255]-> B
    // (invalid lanes hold -HUGE -> __expf gives 0)
    float m = NEG_HUGE;
#pragma unroll
    for (int i = 0; i < 4; ++i) m = fmaxf(m, fmaxf(r[i].x, r[i].y));
    m = wave_max(m);

    float sa = __expf(r[0].x - m) + __expf(r[0].y - m);
    float sb = __expf(r[2].x - m) + __expf(r[2].y - m)
             + __expf(r[3].x - m) + __expf(r[3].y - m);
    {
        float e0 = __expf(r[1].x - m);   // element 64 + 2*lane
        float e1 = __expf(r[1].y - m);   // element 65 + 2*lane
        if (lane <= 18) sa += e0; else sb += e0;   // 64+2*lane < 101
        if (lane <= 17) sa += e1; else sb += e1;   // 65+2*lane < 101
    }
    sa = wave_sum(sa);
    sb = wave_sum(sb);

    if (lane == 0) {
        float ai = (float)accIdx;
        float pa = ai > HALFTOK ? ai / HALFTOK - 1.0f : 1.0f - ai / HALFTOK;
        float c_acc = smooth_l1(pa, gt_acc[row]);

        float si = (float)steerIdx;
        float c_steer = smooth_l1(si / HALFTOK - 1.0f, gt_steer[row]);

        // two-class log-softmax over the *probabilities* (as the reference
        // does): nll = logsumexp(p0,p1) - p_gt
        float tot = sa + sb;
        float p0 = sa / tot, p1 = sb / tot;
        float m2  = fmaxf(p0, p1);
        float lse = m2 + __logf(__expf(p0 - m2) + __expf(p1 - m2));

        int g = gt_rev[row];
        float c_nll = 0.0f, c_valid = 0.0f;
        if (g != IGNORE_IDX) {
            c_nll   = lse - (g ? p1 : p0);
            c_valid = 1.0f;
        }
        s_part[wid][0] = c_acc;
        s_part[wid][1] = c_steer;
        s_part[wid][2] = c_nll;
        s_part[wid][3] = c_valid;
    }
    __syncthreads();

    if (threadIdx.x < 4) {
        float acc = 0.0f;
#pragma unroll
        for (int w = 0; w < WAVES; ++w) acc += s_part[w][threadIdx.x];
        partial[(size_t)blockIdx.x * 4 + threadIdx.x] = acc;
    }
}

// Deterministic final reduce: fixed strided assignment + fixed-order f64 tree.
__global__ __launch_bounds__(256) void control_loss_final(
    const float* __restrict__ partial, float* __restrict__ out)
{
    __shared__ double sd[4][256];
    double a0 = 0, a1 = 0, a2 = 0, a3 = 0;
    for (int i = threadIdx.x; i < NBLK; i += 256) {
        const float* p = partial + (size_t)i * 4;
        a0 += (double)p[0];
        a1 += (double)p[1];
        a2 += (double)p[2];
        a3 += (double)p[3];
    }
    sd[0][threadIdx.x] = a0;
    sd[1][threadIdx.x] = a1;
    sd[2][threadIdx.x] = a2;
    sd[3][threadIdx.x] = a3;
    __syncthreads();
    for (int off = 128; off; off >>= 1) {
        if (threadIdx.x < off) {
#pragma unroll
            for (int k = 0; k < 4; ++k)
                sd[k][threadIdx.x] += sd[k][threadIdx.x + off];
        }
        __syncthreads();
    }
    if (threadIdx.x == 0) {
        const double N = (double)NROWS;
        out[0] = (float)((sd[0][0] + sd[1][0]) / N);            // acc+steer smooth-L1 means
        double vc = sd[3][0] < 1.0 ? 1.0 : sd[3][0];
        out[1] = (float)(sd[2][0] / vc);                        // reverse NLL mean
    }
}

extern "C" void kernel_launch(void* const* d_in, const int* in_sizes, int n_in,
                              void* d_out, int out_size, void* d_ws, size_t ws_size,
                              hipStream_t stream) {
    const float* pred     = (const float*)d_in[0];
    const float* gt_acc   = (const float*)d_in[1];
    const float* gt_steer = (const float*)d_in[2];
    const int*   gt_rev   = (const int*)d_in[3];
    float* partial = (float*)d_ws;   // NBLK*4 floats = 349,184 B of scratch

    control_loss_partial<<<NBLK, BLOCK, 0, stream>>>(pred, gt_acc, gt_steer,
                                                     gt_rev, partial);
    control_loss_final<<<1, 256, 0, stream>>>(partial, (float*)d_out);
}